// VectorQuantizer_59992103190621
// MI455X (gfx1250) — compile-verified
//
#include <hip/hip_runtime.h>
#include <hip/hip_bf16.h>

typedef float v2f __attribute__((ext_vector_type(2)));
typedef float v4f __attribute__((ext_vector_type(4)));
typedef float v8f __attribute__((ext_vector_type(8)));

#define DIM 64
#define NEMB 512
#define NROWS 65536
#define ROWS_PER_BLOCK 128      // 8 waves * 16 rows
#define CHUNK 128               // codes staged in LDS per pass
#define LDSTRIDE 68             // padded dword stride: conflict-free AND 16B-aligned rows
#define ZQ_ELEMS (NROWS * DIM)  // 4194304
#define IDX_ELEMS NROWS         // 65536

__global__ __launch_bounds__(256)
void VectorQuantizer_vq_main(const float* __restrict__ z,
                             const float* __restrict__ ew,
                             float* __restrict__ zq,
                             float* __restrict__ idx_out,
                             float* __restrict__ partial) {
    __shared__ float s_emb[CHUNK * LDSTRIDE];   // 34816 B
    __shared__ float s_enorm[NEMB];             // 2048 B
    __shared__ float s_loss[8];

    const int tid  = threadIdx.x;
    const int lane = tid & 31;
    const int wave = tid >> 5;
    const int half = lane >> 4;   // 0: K lo / rows 0-7 results, 1: K hi / rows 8-15
    const int l16  = lane & 15;
    const int m0   = blockIdx.x * ROWS_PER_BLOCK + wave * 16;
    const int row  = m0 + l16;

    // ---- codebook squared norms (once per block) ----
    for (int c = tid; c < NEMB; c += 256) {
        const float* p = ew + c * DIM;
        float s = 0.f;
        #pragma unroll 16
        for (int k = 0; k < DIM; ++k) s += p[k] * p[k];
        s_enorm[c] = s;
    }

    // ---- A fragments: 16 K-steps of the 16x4 fp32 WMMA layout, kept in VGPRs ----
    v2f a[16];
    #pragma unroll
    for (int k = 0; k < 16; ++k)
        a[k] = *(const v2f*)(z + row * DIM + 4 * k + 2 * half);

    float best_d[8];
    int   best_i[8];
    #pragma unroll
    for (int r = 0; r < 8; ++r) { best_d[r] = 3.4e38f; best_i[r] = 0; }

    for (int cc = 0; cc < NEMB / CHUNK; ++cc) {
        __syncthreads();
        // cooperative stage of 128 codebook rows into LDS, 16B granules
        #pragma unroll
        for (int i = 0; i < (CHUNK * DIM / 4) / 256; ++i) {   // 8 iters
            int idx4 = tid + 256 * i;
            int rr   = idx4 >> 4;    // 0..127
            int c4   = idx4 & 15;    // 0..15
            v4f v = *(const v4f*)(ew + (cc * CHUNK + rr) * DIM + c4 * 4);
            *(v4f*)&s_emb[rr * LDSTRIDE + c4 * 4] = v;
        }
        __syncthreads();

        for (int tt = 0; tt < CHUNK / 16; ++tt) {
            const int e0 = cc * CHUNK + tt * 16;
            const int n  = e0 + l16;             // this lane's code index in tile
            const float en = s_enorm[n];         // issue alongside B-tile loads

            // prefetch the whole B tile for this code block BEFORE any wmma;
            // sched_barrier keeps the 8 ds_load_2addr grouped so waitcnt can
            // emit progressive dscnt waits instead of 8x (load -> wait0 -> wmma)
            v2f b[16];
            #pragma unroll
            for (int k = 0; k < 16; ++k)
                b[k] = *(const v2f*)&s_emb[(tt * 16 + l16) * LDSTRIDE + 4 * k + 2 * half];
            __builtin_amdgcn_sched_barrier(0);

            v8f acc = {};
            #pragma unroll
            for (int k = 0; k < 16; ++k)
                acc = __builtin_amdgcn_wmma_f32_16x16x4_f32(
                    /*neg_a=*/false, a[k], /*neg_b=*/false, b[k],
                    /*c_mod=*/(short)0, acc, /*reuse_a=*/false, /*reuse_b=*/false);

            #pragma unroll
            for (int r = 0; r < 8; ++r) {
                float d = en - 2.0f * acc[r];    // ||e||^2 - 2 z.e  (||z||^2 omitted)
                if (d < best_d[r]) { best_d[r] = d; best_i[r] = n; }
            }
        }
    }

    // ---- argmin across each 16-lane half (lexicographic: first-min on ties) ----
    #pragma unroll
    for (int r = 0; r < 8; ++r) {
        float d = best_d[r]; int i = best_i[r];
        #pragma unroll
        for (int off = 1; off < 16; off <<= 1) {
            float od = __shfl_xor(d, off, 32);
            int   oi = __shfl_xor(i, off, 32);
            if (od < d || (od == d && oi < i)) { d = od; i = oi; }
        }
        best_d[r] = d; best_i[r] = i;
    }

    // ---- write indices (lane 0 -> rows m0..m0+7, lane 16 -> rows m0+8..m0+15) ----
    if (l16 == 0) {
        #pragma unroll
        for (int r = 0; r < 8; ++r)
            idx_out[m0 + half * 8 + r] = (float)best_i[r];
    }

    // ---- z_q gather + commitment-loss partial ----
    float lacc = 0.f;
    #pragma unroll
    for (int m = 0; m < 16; ++m) {
        int bi = __shfl(best_i[m & 7], (m < 8) ? 0 : 16, 32);
        v2f q  = *(const v2f*)(ew + bi * DIM + lane * 2);
        v2f zz = *(const v2f*)(z + (m0 + m) * DIM + lane * 2);
        *(v2f*)(zq + (m0 + m) * DIM + lane * 2) = q;
        float dx = zz.x - q.x, dy = zz.y - q.y;
        lacc += dx * dx + dy * dy;
    }
    #pragma unroll
    for (int off = 16; off >= 1; off >>= 1) lacc += __shfl_xor(lacc, off, 32);
    if (lane == 0) s_loss[wave] = lacc;
    __syncthreads();
    if (tid == 0) {
        float s = 0.f;
        #pragma unroll
        for (int w = 0; w < 8; ++w) s += s_loss[w];   // fixed order: deterministic
        partial[blockIdx.x] = s;
    }
}

__global__ __launch_bounds__(256)
void VectorQuantizer_vq_loss(const float* __restrict__ partial,
                             float* __restrict__ out_loss) {
    __shared__ float s[256];
    const int t = threadIdx.x;
    s[t] = partial[t] + partial[t + 256];             // 512 partials, fixed pairing
    __syncthreads();
    for (int off = 128; off >= 1; off >>= 1) {
        if (t < off) s[t] += s[t + off];
        __syncthreads();
    }
    if (t == 0) out_loss[0] = 0.25f * s[0] / (float)ZQ_ELEMS;
}

extern "C" void kernel_launch(void* const* d_in, const int* in_sizes, int n_in,
                              void* d_out, int out_size, void* d_ws, size_t ws_size,
                              hipStream_t stream) {
    const float* z  = (const float*)d_in[0];   // [64,32,32,64]
    const float* ew = (const float*)d_in[1];   // [512,64]
    float* zq      = (float*)d_out;                       // 4194304 floats
    float* idx_out = (float*)d_out + ZQ_ELEMS;            // 65536 (argmin indices)
    float* loss    = (float*)d_out + ZQ_ELEMS + IDX_ELEMS;// 1
    float* partial = (float*)d_ws;                        // 512 floats scratch

    dim3 grid(NROWS / ROWS_PER_BLOCK);   // 512 blocks
    dim3 block(256);                     // 8 wave32
    VectorQuantizer_vq_main<<<grid, block, 0, stream>>>(z, ew, zq, idx_out, partial);
    VectorQuantizer_vq_loss<<<1, 256, 0, stream>>>(partial, loss);
}